// AudioStructuralAnalyzer_86655260164375
// MI455X (gfx1250) — compile-verified
//
#include <hip/hip_runtime.h>
#include <hip/hip_bf16.h>
#include <stdint.h>

typedef float v2f __attribute__((ext_vector_type(2)));
typedef float v8f __attribute__((ext_vector_type(8)));

#define EPSF 1e-10f
#define FH 256
#define FW 2048
#define TS 32
#define XS_S 41   // 40x40 tile, halo 4, padded stride
#define G3_S 39   // 38x38, halo 3
#define H2_S 37   // 36x36, halo 2
#define TV_S 37   // 32x36 vertical-pass temp
#define SM_S 33   // 32x32 smoothed tile

// normalized 1D gaussian (sigma=1, ws=5); 2D kernel is its outer product
#define GN0 0.05448868454964294f
#define GN1 0.24420134200323348f
#define GN2 0.40261994689424746f

__global__ __launch_bounds__(256)
void audio_struct_kernel(const float* __restrict__ spec, float* __restrict__ out)
{
    __shared__ float xs  [40*XS_S];
    __shared__ float gfS [38*G3_S];
    __shared__ float gtS [38*G3_S];
    __shared__ float uxS [38*G3_S];
    __shared__ float uyS [38*G3_S];
    __shared__ float vx2S[36*H2_S];
    __shared__ float vy2S[36*H2_S];
    __shared__ float vxyS[36*H2_S];
    __shared__ float cvS [36*H2_S];
    __shared__ float tmpv[32*TV_S];
    __shared__ float smS [32*SM_S];

    const int tid = threadIdx.x;
    const int t0 = blockIdx.x * TS;
    const int f0 = blockIdx.y * TS;
    const int b  = blockIdx.z;

    float o[6][4];
    #pragma unroll
    for (int p = 0; p < 6; ++p)
        #pragma unroll
        for (int i = 0; i < 4; ++i) o[p][i] = 0.f;

    const int wv    = tid >> 5;
    const int lane  = tid & 31;
    const int halfW = lane >> 4;
    const int n16   = lane & 15;

    for (int chn = 0; chn < 2; ++chn) {
        const float* sp = spec + (size_t)(b*2 + chn) * FH * FW;

        // ---- Phase A: async global->LDS load of 40x40 haloed tile ----
        for (int idx = tid; idx < 40*40; idx += 256) {
            int r = idx / 40, c = idx - r*40;
            int gr = f0 + r - 4, gc = t0 + c - 4;
            uint32_t la = (uint32_t)(uintptr_t)(&xs[r*XS_S + c]);
            if (gr >= 0 && gr < FH && gc >= 0 && gc < FW) {
                uint64_t ga = (uint64_t)(uintptr_t)(sp + (size_t)gr*FW + gc);
                asm volatile("global_load_async_to_lds_b32 %0, %1, off"
                             :: "v"(la), "v"(ga) : "memory");
            } else {
                xs[r*XS_S + c] = 0.f;
            }
        }
        asm volatile("s_wait_asynccnt 0" ::: "memory");
        __syncthreads();

        // ---- Phase B: Sobel grads + unit orientation (halo 3) ----
        for (int idx = tid; idx < 38*38; idx += 256) {
            int r = idx / 38, c = idx - r*38;
            int gr = f0 + r - 3, gc = t0 + c - 3;
            float gf = 0.f, gt = 0.f, ux = 0.f, uy = 0.f;
            if (gr >= 0 && gr < FH && gc >= 0 && gc < FW) {
                const float* X = &xs[(r+1)*XS_S + (c+1)];
                float a = X[-XS_S-1], bq = X[-XS_S], cq = X[-XS_S+1];
                float d = X[-1],                     e = X[1];
                float f = X[ XS_S-1], g = X[ XS_S], h = X[ XS_S+1];
                gt = (cq - a + 2.f*(e - d) + h - f) * 0.125f;
                gf = (f + 2.f*g + h - a - 2.f*bq - cq) * 0.125f;
                float gte = gt + EPSF;
                float hyp = fmaxf(sqrtf(gf*gf + gte*gte), 1e-30f);
                ux = gte / hyp;
                uy = gf  / hyp;
            }
            gfS[r*G3_S+c] = gf; gtS[r*G3_S+c] = gt;
            uxS[r*G3_S+c] = ux; uyS[r*G3_S+c] = uy;
        }
        __syncthreads();

        // ---- Phase C: structure tensor + curvature magnitude (halo 2) ----
        for (int idx = tid; idx < 36*36; idx += 256) {
            int r = idx / 36, c = idx - r*36;
            int gr = f0 + r - 2, gc = t0 + c - 2;
            float a = 0.f, bq = 0.f, cq = 0.f, cv = 0.f;
            if (gr >= 0 && gr < FH && gc >= 0 && gc < FW) {
                int i3 = (r+1)*G3_S + (c+1);
                float gf = gfS[i3], gt = gtS[i3], ux = uxS[i3], uy = uyS[i3];
                float mag2 = gf*gf + gt*gt + EPSF;
                a  = mag2*ux*ux; bq = mag2*uy*uy; cq = mag2*ux*uy;
                const float* U = &uxS[i3];
                const float* V = &uyS[i3];
                float dudx = (U[-G3_S+1]-U[-G3_S-1] + 2.f*(U[1]-U[-1]) + U[G3_S+1]-U[G3_S-1])*0.125f;
                float dudy = (U[G3_S-1]+2.f*U[G3_S]+U[G3_S+1] - U[-G3_S-1]-2.f*U[-G3_S]-U[-G3_S+1])*0.125f;
                float dvdx = (V[-G3_S+1]-V[-G3_S-1] + 2.f*(V[1]-V[-1]) + V[G3_S+1]-V[G3_S-1])*0.125f;
                float dvdy = (V[G3_S-1]+2.f*V[G3_S]+V[G3_S+1] - V[-G3_S-1]-2.f*V[-G3_S]-V[-G3_S+1])*0.125f;
                cv = sqrtf(dudx*dudx + dudy*dudy + dvdx*dvdx + dvdy*dvdy + EPSF);
            }
            int i2 = r*H2_S + c;
            vx2S[i2] = a; vy2S[i2] = bq; vxyS[i2] = cq; cvS[i2] = cv;
        }
        __syncthreads();

        float sv[9][4];

        // ---- Phase D: 9 separable 5x5 gaussians; vertical=VALU, horizontal=WMMA ----
        #pragma unroll
        for (int fld = 0; fld < 9; ++fld) {
            // vertical 5-tap into tmpv (32 rows x 36 cols, col rel = cc-2)
            for (int idx = tid; idx < 32*36; idx += 256) {
                int r = idx / 36, cc = idx - r*36;
                int gc = t0 + cc - 2;
                float acc = 0.f;
                #pragma unroll
                for (int k = 0; k < 5; ++k) {
                    int rr = r + k - 2;
                    int gr = f0 + rr;
                    float v = 0.f;
                    if (gr >= 0 && gr < FH && gc >= 0 && gc < FW) {
                        if      (fld == 0) v = vx2S[(rr+2)*H2_S + cc];
                        else if (fld == 1) v = vy2S[(rr+2)*H2_S + cc];
                        else if (fld == 2) v = vxyS[(rr+2)*H2_S + cc];
                        else if (fld == 3) v = uxS[(rr+3)*G3_S + cc + 1];
                        else if (fld == 4) v = uyS[(rr+3)*G3_S + cc + 1];
                        else if (fld == 5) v = cvS[(rr+2)*H2_S + cc];
                        else if (fld == 6) { float xv = xs[(rr+4)*XS_S + cc + 2]; v = xv*xv; }
                        else if (fld == 7) { float g = gtS[(rr+3)*G3_S + cc + 1]; v = 1.f/(1.f + fabsf(g)); }
                        else               v = fabsf(gfS[(rr+3)*G3_S + cc + 1]);
                    }
                    const float gk = (k==0||k==4) ? GN0 : ((k==2) ? GN2 : GN1);
                    acc += gk * v;
                }
                tmpv[r*TV_S + cc] = acc;
            }
            __syncthreads();

            // horizontal 5-tap as T(16x20) x W(20x16) = five WMMA f32 16x16x4
            if (wv < 4) {
                const int tr = (wv >> 1) * 16;
                const int tc = (wv & 1) * 16;
                v8f acc = {0.f,0.f,0.f,0.f,0.f,0.f,0.f,0.f};
                #pragma unroll
                for (int chnk = 0; chnk < 5; ++chnk) {
                    v2f av, bv;
                    // A layout: M = lane&15, K = v + 2*(lane>>4)  (global K += 4*chnk)
                    int base = (tr + n16)*TV_S + tc + 4*chnk + 2*halfW;
                    av.x = tmpv[base];
                    av.y = tmpv[base + 1];
                    // B layout (mirror): K = v + 2*half, N = lane&15; banded W[K,N]=gn[N+2-K]
                    #pragma unroll
                    for (int vvi = 0; vvi < 2; ++vvi) {
                        int kk = 4*chnk + vvi + 2*halfW;
                        int d = n16 + 2 - kk;
                        float w = (d==0||d==4) ? GN0 : ((d==1||d==3) ? GN1 : ((d==2) ? GN2 : 0.f));
                        if (vvi == 0) bv.x = w; else bv.y = w;
                    }
                    acc = __builtin_amdgcn_wmma_f32_16x16x4_f32(
                        false, av, false, bv, (short)0, acc, false, false);
                }
                // D layout: M = v + 8*half, N = lane&15
                #pragma unroll
                for (int vvi = 0; vvi < 8; ++vvi) {
                    int m = vvi + 8*halfW;
                    smS[(tr + m)*SM_S + tc + n16] = acc[vvi];
                }
            }
            __syncthreads();

            #pragma unroll
            for (int i = 0; i < 4; ++i) {
                int lin = tid + 256*i;
                int r = lin >> 5, c = lin & 31;
                sv[fld][i] = smS[r*SM_S + c];
            }
        }

        // ---- Phase E: pointwise combine + channel accumulation ----
        #pragma unroll
        for (int i = 0; i < 4; ++i) {
            int lin = tid + 256*i;
            int r = lin >> 5, c = lin & 31;
            float vxx = sv[0][i], vyy = sv[1][i], vxy = sv[2][i];
            float uxa = sv[3][i], uya = sv[4][i];
            float cur = sv[5][i], le = sv[6][i], tmS = sv[7][i], spS = sv[8][i];

            float trace = vxx + vyy;
            float det   = vxx*vyy - vxy*vxy;
            float disc  = sqrtf(fmaxf(trace*trace - 4.f*det, 0.f) + EPSF);
            float l1 = fmaxf((trace + disc)*0.5f, EPSF);
            float l2 = fmaxf((trace - disc)*0.5f, EPSF);
            float s  = l1 + l2 + EPSF;
            float p1 = l1/s, p2 = l2/s;
            float ent = -(p1*logf(p1+EPSF) + p2*logf(p2+EPSF)) * 1.4426950408889634f;
            ent = fminf(fmaxf(ent, 0.f), 1.f);

            float alg = fminf(fmaxf(sqrtf(uxa*uxa + uya*uya + EPSF), 0.f), 1.f);

            float xm3 = xs[(r+1)*XS_S + c + 4];
            float x0  = xs[(r+4)*XS_S + c + 4];
            float xp3 = xs[(r+7)*XS_S + c + 4];
            float hc  = fminf(fmaxf(fabsf(2.f*x0 - xm3 - xp3) / (le + EPSF), 0.f), 1.f);

            float tem = fminf(fmaxf(tmS, 0.f), 1.f);
            float spe = fminf(fmaxf(spS, 0.f), 1.f);

            o[0][i] += ent; o[1][i] += alg; o[2][i] += cur;
            o[3][i] += hc;  o[4][i] += tem; o[5][i] += spe;
        }
        __syncthreads();   // xs reused next channel
    }

    const size_t PLANE = (size_t)8 * FH * FW;
    const size_t IMG   = (size_t)FH * FW;
    #pragma unroll
    for (int i = 0; i < 4; ++i) {
        int lin = tid + 256*i;
        int r = lin >> 5, c = lin & 31;
        size_t off = (size_t)b*IMG + (size_t)(f0 + r)*FW + (size_t)(t0 + c);
        #pragma unroll
        for (int p = 0; p < 6; ++p)
            out[(size_t)p*PLANE + off] = o[p][i] * 0.5f;
    }
}

extern "C" void kernel_launch(void* const* d_in, const int* in_sizes, int n_in,
                              void* d_out, int out_size, void* d_ws, size_t ws_size,
                              hipStream_t stream) {
    const float* spec = (const float*)d_in[0];
    float* out = (float*)d_out;
    dim3 grid(FW / TS, FH / TS, 8);   // 64 x 8 x 8 = 4096 blocks
    dim3 block(256);
    audio_struct_kernel<<<grid, block, 0, stream>>>(spec, out);
}